// BasicModel_63848983823018
// MI455X (gfx1250) — compile-verified
//
#include <hip/hip_runtime.h>
#include <hip/hip_bf16.h>

typedef __attribute__((ext_vector_type(8)))  float    v8f;
typedef __attribute__((ext_vector_type(2)))  float    f32x2;

#define NQ     5
#define DIM    32
#define NBLK   8
#define NCOPY  16

// ---- gate helpers: lane = amplitude index s (0..31), qubit q <-> bit (4-q) ----

template<int NC>
__device__ __forceinline__ void apply_ry(float (&re)[NCOPY], float (&im)[NCOPY],
                                         const float (&th)[NQ], int lane) {
  for (int q = 0; q < NQ; ++q) {
    float hs, hc;
    __sincosf(0.5f * th[q], &hs, &hc);
    const int mask = 1 << (4 - q);
    const float sg = (lane & mask) ? hs : -hs;   // bit=1: +s*partner, bit=0: -s*partner
#pragma unroll
    for (int j = 0; j < NC; ++j) {
      float pr = re[j], pi = im[j];
      float qr = __shfl_xor(pr, mask, 32);
      float qi = __shfl_xor(pi, mask, 32);
      re[j] = hc * pr + sg * qr;
      im[j] = hc * pi + sg * qi;
    }
  }
}

template<int NC>
__device__ __forceinline__ void apply_rz(float (&re)[NCOPY], float (&im)[NCOPY],
                                         const float (&th)[NQ], int lane) {
  float phi = 0.0f;
#pragma unroll
  for (int q = 0; q < NQ; ++q)
    phi += ((lane >> (4 - q)) & 1) ? 0.5f * th[q] : -0.5f * th[q];
  float sp, cp;
  __sincosf(phi, &sp, &cp);
#pragma unroll
  for (int j = 0; j < NC; ++j) {
    float r = re[j], i = im[j];
    re[j] = r * cp - i * sp;
    im[j] = r * sp + i * cp;
  }
}

// state[s] <- state[TAU_L[s]]; TAU_L[k] = s0(s1(s2(s3(s4(k))))),
// s_i(k) = k ^ (((k >> (4-i)) & 1) << (4 - ((i+L)%5)))
template<int NC>
__device__ __forceinline__ void apply_tau(float (&re)[NCOPY], float (&im)[NCOPY],
                                          int L, int lane) {
  int v = lane;
#pragma unroll
  for (int i = 4; i >= 0; --i) {
    int t = (i + L) % 5;
    int bit = (v >> (4 - i)) & 1;
    v ^= bit << (4 - t);
  }
#pragma unroll
  for (int j = 0; j < NC; ++j) {
    re[j] = __shfl(re[j], v, 32);
    im[j] = __shfl(im[j], v, 32);
  }
}

__global__ __launch_bounds__(256)
void qcircuit_fused(const float* __restrict__ x,
                    const float* __restrict__ w0,  const float* __restrict__ w1a,
                    const float* __restrict__ w1b, const float* __restrict__ w1c,
                    const float* __restrict__ w2,  const float* __restrict__ w3a,
                    const float* __restrict__ w3b, const float* __restrict__ w3c,
                    float* __restrict__ outState, float* __restrict__ outO) {
  const int b    = blockIdx.x;
  const int n    = threadIdx.x >> 5;   // block index within sample (wave id)
  const int lane = threadIdx.x & 31;   // amplitude index s

  // ---- load angles (lane-uniform broadcasts) ----
  float thX[NQ], tw0[NQ], t1a[NQ], t1b[NQ], t1c[NQ], tw2[NQ], t3a[NQ], t3b[NQ], t3c[NQ];
#pragma unroll
  for (int q = 0; q < NQ; ++q) {
    const int wi = n * NQ + q;
    thX[q] = x[(size_t)b * (NBLK * NQ) + wi];
    tw0[q] = w0[wi];  t1a[q] = w1a[wi]; t1b[q] = w1b[wi]; t1c[q] = w1c[wi];
    tw2[q] = w2[wi];  t3a[q] = w3a[wi]; t3b[q] = w3b[wi]; t3c[q] = w3c[wi];
  }

  // ---- evolve |0> ----
  float re[NCOPY], im[NCOPY];
  re[0] = (lane == 0) ? 1.0f : 0.0f;
  im[0] = 0.0f;

  apply_ry<1>(re, im, thX, lane);
  apply_ry<1>(re, im, tw0, lane);
  apply_ry<1>(re, im, thX, lane);
  apply_rz<1>(re, im, t1a, lane);
  apply_ry<1>(re, im, t1b, lane);
  apply_rz<1>(re, im, t1c, lane);
  apply_tau<1>(re, im, 1, lane);
  apply_ry<1>(re, im, thX, lane);

  // ---- expand to 16 signed copies; ent-factor sign mask per (n, s) ----
  unsigned nm = 0u;
  if (n < 2) {
    nm |= ((lane >> 4) & 1) << 0;                              // g0: q=0 -> bit4
    nm |= (lane & 1) << 1;                                     // g1: q=4 -> bit0
    nm |= (((n == 0) ? (lane >> 3) : (lane >> 1)) & 1) << 2;   // g2: q=1/q=3
    nm |= ((lane >> 2) & 1) << 3;                              // g3: q=2 -> bit2
  }
  {
    const float r0v = re[0], i0v = im[0];
#pragma unroll
    for (int j = 0; j < NCOPY; ++j) {
      const float sg = (__popc(j & (int)nm) & 1) ? -1.0f : 1.0f;
      re[j] = sg * r0v;
      im[j] = sg * i0v;
    }
  }

  apply_ry<NCOPY>(re, im, tw2, lane);
  apply_ry<NCOPY>(re, im, thX, lane);
  apply_rz<NCOPY>(re, im, t3a, lane);
  apply_ry<NCOPY>(re, im, t3b, lane);
  apply_rz<NCOPY>(re, im, t3c, lane);
  apply_tau<NCOPY>(re, im, 2, lane);

  // ---- stream final state (complex64 -> float pairs), non-temporal ----
  {
    f32x2* outc = (f32x2*)outState + ((size_t)(b * NBLK + n) * NCOPY) * DIM;
#pragma unroll
    for (int j = 0; j < NCOPY; ++j) {
      f32x2 v; v.x = re[j]; v.y = im[j];
      __builtin_nontemporal_store(v, outc + j * DIM + lane);
    }
  }

  // ---- stage f32 state for WMMA Gram matrix M = S* diag(OBS) S^T ----
  __shared__ alignas(64) float sRe[NBLK][NCOPY][DIM];
  __shared__ alignas(64) float sIm[NBLK][NCOPY][DIM];
  __shared__ alignas(16) float sM[NBLK][NCOPY][NCOPY][2];
  __shared__ float sRed[NBLK];

#pragma unroll
  for (int j = 0; j < NCOPY; ++j) {
    sRe[n][j][lane] = re[j];
    sIm[n][j][lane] = im[j];
  }
  __syncthreads();

  const int mrow = lane & 15;   // A row (copy j) == B column (copy k)
  const int hi   = lane >> 4;

  // V_WMMA_F32_16X16X4_F32, K accumulated over 8 chunks of 4.
  //   A[m,K]   : lanes<16 hold K = 4c+0,4c+1 ; lanes>=16 hold K = 4c+2,4c+3
  //   B[K,ncol]: same K split per lane half -> SAME LDS addresses as A,
  //              with OBS[K] = (-1)^popc(K) folded in. popc(4c+2hi+e) = popc(c)+hi+e,
  //              so chunk sign s0 = (-1)^(popc(c)+hi), and element .y is -s0.
  v8f accRe  = {};
  v8f accImP = {};   // sum_K A_R * B_I
  v8f accImN = {};   // sum_K A_I * B_R   (subtract at end)
#pragma unroll
  for (int c = 0; c < 8; ++c) {
    f32x2 pR = *(const f32x2*)&sRe[n][mrow][4 * c + 2 * hi];
    f32x2 pI = *(const f32x2*)&sIm[n][mrow][4 * c + 2 * hi];
    const float s0 = (((__popc(c) & 1) ^ hi) ? -1.0f : 1.0f);
    f32x2 bR; bR.x = s0 * pR.x; bR.y = -s0 * pR.y;
    f32x2 bI; bI.x = s0 * pI.x; bI.y = -s0 * pI.y;
    accRe  = __builtin_amdgcn_wmma_f32_16x16x4_f32(false, pR, false, bR, (short)0, accRe,  false, false);
    accRe  = __builtin_amdgcn_wmma_f32_16x16x4_f32(false, pI, false, bI, (short)0, accRe,  false, false);
    accImP = __builtin_amdgcn_wmma_f32_16x16x4_f32(false, pR, false, bI, (short)0, accImP, false, false);
    accImN = __builtin_amdgcn_wmma_f32_16x16x4_f32(false, pI, false, bR, (short)0, accImN, false, false);
  }

  // C layout: VGPR r -> row (r + 8*hi), col = lane&15
#pragma unroll
  for (int r = 0; r < 8; ++r) {
    const int mr = r + 8 * hi;
    f32x2 mv; mv.x = accRe[r]; mv.y = accImP[r] - accImN[r];
    *(f32x2*)&sM[n][mr][mrow][0] = mv;
  }
  __syncthreads();

  // ---- P = prod_n M_n (elementwise complex), O = Re( sum conj(a_j) a_k P[j,k] ) ----
  {
    const int t  = threadIdx.x;
    const int mj = t >> 4;
    const int mk = t & 15;
    float pr = 1.0f, pi = 0.0f;
#pragma unroll
    for (int nn = 0; nn < NBLK; ++nn) {
      const float ar = sM[nn][mj][mk][0];
      const float ai = sM[nn][mj][mk][1];
      const float nr = pr * ar - pi * ai;
      const float ni = pr * ai + pi * ar;
      pr = nr; pi = ni;
    }
    // conj(alpha_j)*alpha_k = (1/16) * i^((pop(j)-pop(k)) mod 4)
    const int d = (__popc(mj) - __popc(mk)) & 3;
    float val = (d == 0) ? pr : (d == 1) ? -pi : (d == 2) ? -pr : pi;
    val *= 0.0625f;

#pragma unroll
    for (int off = 16; off > 0; off >>= 1) val += __shfl_down(val, off, 32);
    if (lane == 0) sRed[n] = val;
    __syncthreads();
    if (threadIdx.x == 0) {
      float o = 0.0f;
#pragma unroll
      for (int i2 = 0; i2 < NBLK; ++i2) o += sRed[i2];
      outO[b] = o;
    }
  }
}

extern "C" void kernel_launch(void* const* d_in, const int* in_sizes, int n_in,
                              void* d_out, int out_size, void* d_ws, size_t ws_size,
                              hipStream_t stream) {
  (void)n_in; (void)out_size; (void)d_ws; (void)ws_size;
  const float* x   = (const float*)d_in[0];
  const float* w0  = (const float*)d_in[1];
  const float* w1a = (const float*)d_in[2];
  const float* w1b = (const float*)d_in[3];
  const float* w1c = (const float*)d_in[4];
  const float* w2  = (const float*)d_in[5];
  const float* w3a = (const float*)d_in[6];
  const float* w3b = (const float*)d_in[7];
  const float* w3c = (const float*)d_in[8];

  const int B = in_sizes[0] / (NBLK * NQ);           // 4096
  float* outState = (float*)d_out;                   // B*8*16*32 complex64 -> float pairs
  float* outO     = outState + (size_t)B * NBLK * NCOPY * DIM * 2;

  qcircuit_fused<<<dim3(B), dim3(256), 0, stream>>>(
      x, w0, w1a, w1b, w1c, w2, w3a, w3b, w3c, outState, outO);
}